// GraphConv_19997367730723
// MI455X (gfx1250) — compile-verified
//
#include <hip/hip_runtime.h>

#define D 256
#define GAMMAF 0.2f
#define EPSF 1e-8f

typedef __attribute__((ext_vector_type(2))) float v2f;
typedef __attribute__((ext_vector_type(8))) float v8f;

__device__ __forceinline__ float cleanf(float x) {
  if (x != x) return 0.0f;                       // NaN -> 0
  if (x == __builtin_inff()) return 10000.0f;    // +inf -> 1e4
  if (x == -__builtin_inff()) return 0.0001f;    // -inf -> 1e-4
  return x;
}

// ---- per-edge alpha/eta precompute -----------------------------------------

__global__ void edge_stats_kernel(const int* __restrict__ head,
                                  const float* __restrict__ omega,
                                  float* __restrict__ hsum,
                                  float* __restrict__ cntv, int E) {
  int e = blockIdx.x * blockDim.x + threadIdx.x;
  if (e >= E) return;
  int h = head[e];
  unsafeAtomicAdd(&hsum[h], omega[e]);
  unsafeAtomicAdd(&cntv[h], 1.0f);
}

__global__ void alpha_eta0_kernel(const int* __restrict__ head,
                                  const float* __restrict__ omega,
                                  const float* __restrict__ hsum,
                                  float* __restrict__ alpha,
                                  float* __restrict__ eta,
                                  float* __restrict__ esum, int E) {
  int e = blockIdx.x * blockDim.x + threadIdx.x;
  if (e >= E) return;
  int h = head[e];
  float a = omega[e] / (hsum[h] + EPSF);
  alpha[e] = a;
  float e0 = (a > GAMMAF) ? a : 0.0f;
  eta[e] = e0;
  unsafeAtomicAdd(&esum[h], e0);
}

__global__ void eta_fin_kernel(const int* __restrict__ head,
                               const float* __restrict__ esum,
                               float* __restrict__ eta, int E) {
  int e = blockIdx.x * blockDim.x + threadIdx.x;
  if (e >= E) return;
  eta[e] = eta[e] / (esum[head[e]] + EPSF);
}

// ---- scatter kernels: one wave32 per edge, lane owns 8 contiguous floats ---

__global__ void kg_scatter_kernel(const int* __restrict__ head,
                                  const int* __restrict__ tail,
                                  const int* __restrict__ etype,
                                  const float* __restrict__ rho,
                                  const float* __restrict__ rel,
                                  const float* __restrict__ ent,
                                  float* __restrict__ agg, int E) {
  int w = blockIdx.x * (blockDim.x >> 5) + (threadIdx.x >> 5);
  int lane = threadIdx.x & 31;
  if (w >= E) return;
  int h = head[w], t = tail[w], r = etype[w] - 1;
  float rv = rho[w];
  const float4* trow = (const float4*)(ent + (size_t)t * D);
  const float4* rrow = (const float4*)(rel + (size_t)r * D);
  float* drow = agg + (size_t)h * D;
  int c0 = lane * 2;
  float4 t0 = trow[c0], t1 = trow[c0 + 1];
  float4 r0 = rrow[c0], r1 = rrow[c0 + 1];
  int b = lane * 8;
  unsafeAtomicAdd(&drow[b + 0], rv * t0.x * r0.x);
  unsafeAtomicAdd(&drow[b + 1], rv * t0.y * r0.y);
  unsafeAtomicAdd(&drow[b + 2], rv * t0.z * r0.z);
  unsafeAtomicAdd(&drow[b + 3], rv * t0.w * r0.w);
  unsafeAtomicAdd(&drow[b + 4], rv * t1.x * r1.x);
  unsafeAtomicAdd(&drow[b + 5], rv * t1.y * r1.y);
  unsafeAtomicAdd(&drow[b + 6], rv * t1.z * r1.z);
  unsafeAtomicAdd(&drow[b + 7], rv * t1.w * r1.w);
}

__global__ void usr_scatter_kernel(const int* __restrict__ iu,
                                   const int* __restrict__ ii,
                                   const float* __restrict__ wv,
                                   const float* __restrict__ ent,
                                   float* __restrict__ agg, int IE) {
  int w = blockIdx.x * (blockDim.x >> 5) + (threadIdx.x >> 5);
  int lane = threadIdx.x & 31;
  if (w >= IE) return;
  int u = iu[w], it = ii[w];
  float ww = wv[w];
  const float4* trow = (const float4*)(ent + (size_t)it * D);
  float* drow = agg + (size_t)u * D;
  int c0 = lane * 2;
  float4 t0 = trow[c0], t1 = trow[c0 + 1];
  int b = lane * 8;
  unsafeAtomicAdd(&drow[b + 0], ww * t0.x);
  unsafeAtomicAdd(&drow[b + 1], ww * t0.y);
  unsafeAtomicAdd(&drow[b + 2], ww * t0.z);
  unsafeAtomicAdd(&drow[b + 3], ww * t0.w);
  unsafeAtomicAdd(&drow[b + 4], ww * t1.x);
  unsafeAtomicAdd(&drow[b + 5], ww * t1.y);
  unsafeAtomicAdd(&drow[b + 6], ww * t1.z);
  unsafeAtomicAdd(&drow[b + 7], ww * t1.w);
}

// ---- normalize: 16-row tile per wave; sum-of-squares via WMMA Gram diag ----
// A layout (16x4 f32): lane m (0..15) holds row m, VGPR0/1 = K0/K1; lanes
// 16..31 hold K2/K3. Using the same registers for B yields B = A^T, so the
// accumulated 16x16 C is the Gram matrix; its diagonal is per-row sum-sq.
// C layout: C[m][n] at VGPR (m&7), lane (n + 16*(m>=8)).

__global__ void normalize_kernel(const float* __restrict__ agg,
                                 const float* __restrict__ cntv,  // null => 1
                                 float* __restrict__ cur,         // may be null
                                 float* __restrict__ res,
                                 int ntiles) {
  __shared__ float lnorm[8 * 256];
  int wid  = threadIdx.x >> 5;
  int lane = threadIdx.x & 31;
  int tile = blockIdx.x * (blockDim.x >> 5) + wid;
  bool active = tile < ntiles;     // wave-uniform -> EXEC all-ones inside
  float* wl = &lnorm[wid * 256];

  if (active) {
    int r0 = tile * 16;
    int m  = lane & 15;
    int hv = lane >> 4;            // half select: K pair 0/1 vs 2/3
    int row = r0 + m;
    float inv = 1.0f;
    if (cntv) inv = 1.0f / fmaxf(cntv[row], 1.0f);
    const v2f* row2 = (const v2f*)(agg + (size_t)row * D);
    v8f c = {};
#pragma unroll 8
    for (int kb = 0; kb < D; kb += 4) {
      v2f a = row2[(kb >> 1) + hv];
      a.x *= inv;
      a.y *= inv;
      c = __builtin_amdgcn_wmma_f32_16x16x4_f32(false, a, false, a,
                                                (short)0, c, false, false);
    }
#pragma unroll
    for (int i = 0; i < 8; ++i) wl[lane * 8 + i] = c[i];
  }
  __syncthreads();
  if (active) {
    int r0 = tile * 16;
    for (int r = 0; r < 16; ++r) {
      int row = r0 + r;
      // diag (r,r): owner lane r / vgpr r (r<8), else lane r+16 / vgpr r-8
      float ss = wl[(r < 8) ? (9 * r) : (9 * r + 120)];
      float inv = 1.0f;
      if (cntv) inv = 1.0f / fmaxf(cntv[row], 1.0f);
      float scale = inv / fmaxf(sqrtf(ss), EPSF);
      const float4* arow = (const float4*)(agg + (size_t)row * D);
      float4 v0 = arow[lane * 2];
      float4 v1 = arow[lane * 2 + 1];
      float o[8] = {v0.x, v0.y, v0.z, v0.w, v1.x, v1.y, v1.z, v1.w};
      float* rr = res + (size_t)row * D + lane * 8;
      float* cr = cur ? (cur + (size_t)row * D + lane * 8) : nullptr;
#pragma unroll
      for (int i = 0; i < 8; ++i) {
        float vvv = cleanf(o[i] * scale);
        rr[i] += vvv;
        if (cr) cr[i] = vvv;
      }
    }
  }
}

// ---------------------------------------------------------------------------

extern "C" void kernel_launch(void* const* d_in, const int* in_sizes, int n_in,
                              void* d_out, int out_size, void* d_ws, size_t ws_size,
                              hipStream_t stream) {
  const float* user_emb   = (const float*)d_in[0];
  const float* entity_emb = (const float*)d_in[1];
  const int*   edge_index = (const int*)d_in[2];
  const int*   edge_type  = (const int*)d_in[3];
  const float* omega      = (const float*)d_in[4];
  const int*   inter_edge = (const int*)d_in[5];
  const float* inter_w    = (const float*)d_in[6];
  const float* rel_emb    = (const float*)d_in[7];

  const int E  = in_sizes[3];
  const int IE = in_sizes[6];
  const int n_users = in_sizes[0] / D;
  const int n_ent   = in_sizes[1] / D;

  const int* head = edge_index;
  const int* tail = edge_index + E;
  const int* iu = inter_edge;
  const int* ii = inter_edge + IE;

  float* ws = (float*)d_ws;
  size_t off = 0;
  float* hsum    = ws + off; off += n_ent;
  float* esum    = ws + off; off += n_ent;
  float* cntv    = ws + off; off += n_ent;
  float* alpha   = ws + off; off += E;
  float* eta     = ws + off; off += E;
  float* ent_cur = ws + off; off += (size_t)n_ent * D;
  float* ent_agg = ws + off; off += (size_t)n_ent * D;
  float* usr_agg = ws + off; off += (size_t)n_users * D;
  (void)ws_size; (void)n_in; (void)out_size;

  float* out_ent = (float*)d_out;
  float* out_usr = out_ent + (size_t)n_ent * D;

  // --- edge weights (alpha / eta), edge counts ---
  hipMemsetAsync(hsum, 0, sizeof(float) * (size_t)3 * n_ent, stream);  // hsum,esum,cntv
  const int tpb = 256;
  edge_stats_kernel<<<(E + tpb - 1) / tpb, tpb, 0, stream>>>(head, omega, hsum, cntv, E);
  alpha_eta0_kernel<<<(E + tpb - 1) / tpb, tpb, 0, stream>>>(head, omega, hsum, alpha, eta, esum, E);
  eta_fin_kernel<<<(E + tpb - 1) / tpb, tpb, 0, stream>>>(head, esum, eta, E);

  // --- init current entity state and residual outputs ---
  hipMemcpyAsync(ent_cur, entity_emb, sizeof(float) * (size_t)n_ent * D,
                 hipMemcpyDeviceToDevice, stream);
  hipMemcpyAsync(out_ent, entity_emb, sizeof(float) * (size_t)n_ent * D,
                 hipMemcpyDeviceToDevice, stream);
  hipMemcpyAsync(out_usr, user_emb, sizeof(float) * (size_t)n_users * D,
                 hipMemcpyDeviceToDevice, stream);

  const int wpb = 8;  // waves per 256-thread block
  const int et = n_ent / 16, ut = n_users / 16;
  for (int hop = 1; hop <= 3; ++hop) {
    // ent_agg and usr_agg are contiguous -> single memset
    hipMemsetAsync(ent_agg, 0, sizeof(float) * ((size_t)n_ent + (size_t)n_users) * D, stream);
    const float* rho = (hop < 3) ? alpha : eta;
    kg_scatter_kernel<<<(E + wpb - 1) / wpb, 256, 0, stream>>>(
        head, tail, edge_type, rho, rel_emb, ent_cur, ent_agg, E);
    usr_scatter_kernel<<<(IE + wpb - 1) / wpb, 256, 0, stream>>>(
        iu, ii, inter_w, ent_cur, usr_agg, IE);
    normalize_kernel<<<(et + 7) / 8, 256, 0, stream>>>(ent_agg, cntv, ent_cur, out_ent, et);
    normalize_kernel<<<(ut + 7) / 8, 256, 0, stream>>>(usr_agg, nullptr, nullptr, out_usr, ut);
  }
}